// ToBEVConvolution_43817256354330
// MI455X (gfx1250) — compile-verified
//
#include <hip/hip_runtime.h>
#include <hip/hip_bf16.h>
#include <cstdint>

#define C_IN 16
#define C_OUT 32
#define NKER 64
#define COORD_MAX 64
#define KEYS (COORD_MAX * COORD_MAX * COORD_MAX) /* 262144 */

typedef __attribute__((ext_vector_type(2))) float v2f;
typedef __attribute__((ext_vector_type(4))) float v4f;
typedef __attribute__((ext_vector_type(8))) float v8f;
typedef __attribute__((ext_vector_type(4))) int   v4i;

// CDNA5 async copy: global -> LDS, per-lane LDS destination (lets us fuse the
// transpose into the DMA).  Tracked by ASYNCcnt.
__device__ __forceinline__ void async_ld_lds_b32(unsigned lds_off,
                                                 const float* gptr) {
    asm volatile("global_load_async_to_lds_b32 %0, %1, off"
                 :: "v"(lds_off), "v"(gptr) : "memory");
}
__device__ __forceinline__ void wait_asynccnt0() {
    asm volatile("s_wait_asynccnt 0x0" ::: "memory");
}

// ---------------------------------------------------------------------------
// Kernel 0: zero the dense L2-resident accumulator grid + flags, prefill the
// output (feats=0, coords=-1) so padding slots match the reference.
// ---------------------------------------------------------------------------
__global__ __launch_bounds__(256) void zero_init_kernel(
    float* __restrict__ acc_grid, int* __restrict__ flags,
    float* __restrict__ outf, int* __restrict__ outc, long long n)
{
    const long long gridElems  = (long long)KEYS * C_OUT;
    const long long featElems  = n * C_OUT;
    const long long coordElems = n * 4;
    const long long total = gridElems + KEYS + featElems + coordElems;
    const long long stride = (long long)gridDim.x * blockDim.x;
    for (long long i = (long long)blockIdx.x * blockDim.x + threadIdx.x;
         i < total; i += stride) {
        if (i < gridElems) {
            acc_grid[i] = 0.0f;
        } else if (i < gridElems + KEYS) {
            flags[i - gridElems] = 0;
        } else if (i < gridElems + KEYS + featElems) {
            outf[i - gridElems - KEYS] = 0.0f;
        } else {
            outc[i - gridElems - KEYS - featElems] = -1;
        }
    }
}

// ---------------------------------------------------------------------------
// Kernel 1: per-16-point tile WMMA GEMV (f32 16x16x4) with per-point kernel
// selection via ballot/readlane distinct-k loop, then atomic scatter into the
// dense accumulator grid (L2-resident).
//
// A (16x4 f32): lanes 0-15 = rows, VGPR0 = K(0|2), VGPR1 = K(1|3) per half.
// B (4x16 f32): lanes = columns(o), VGPR0 = K(0|2), VGPR1 = K(1|3) per half.
// C/D (16x16 f32): VGPR g = rows g (lanes 0-15) and g+8 (lanes 16-31).
// ---------------------------------------------------------------------------
__global__ __launch_bounds__(256) void bev_gemv_scatter(
    const float* __restrict__ feat, const int* __restrict__ coords,
    const float* __restrict__ W, const int* __restrict__ strideP,
    float* __restrict__ acc_grid, int* __restrict__ flags, int n)
{
    // Full kernel bank, transposed to [k][o][i] so (i, i+1) pairs are
    // contiguous for ds_load_b64.  64*32*16 f32 = 128 KB of the WGP's 320 KB.
    // Filled with transposing async global->LDS DMA (per-lane LDS addresses).
    __shared__ float sWT[NKER * C_OUT * C_IN];
    {
        const unsigned base = (unsigned)(uintptr_t)&sWT[0];
        for (int f = threadIdx.x; f < NKER * C_IN * C_OUT; f += blockDim.x) {
            int k = f >> 9;        // / (16*32)
            int r = f & 511;
            int i = r >> 5;        // / 32
            int o = r & 31;
            async_ld_lds_b32(base + 4u * (unsigned)(k * 512 + o * 16 + i),
                             W + f);
        }
        wait_asynccnt0();
    }
    __syncthreads();

    const int strideV = strideP[0];
    const int lane = threadIdx.x & 31;
    const int row  = lane & 15;      // A row / C row-low
    const int hi   = lane >> 4;      // lane-half selector
    const int n16  = lane & 15;      // B/C column within a 16-wide half

    const int nTiles = (n + 15) >> 4;
    const int waveId = blockIdx.x * (blockDim.x >> 5) + (threadIdx.x >> 5);
    const int nWaves = gridDim.x * (blockDim.x >> 5);

    for (int t = waveId; t < nTiles; t += nWaves) {   // wave-uniform loop
        int p = t * 16 + row;
        int valid = (p < n) ? 1 : 0;
        int pc = valid ? p : (n - 1);

        v4i cd = *(const v4i*)(coords + (size_t)pc * 4);   // one b128 load
        int myk   = cd.y / strideV;                        // kernel index
        int mykey = cd.x * (COORD_MAX * COORD_MAX) + cd.z * COORD_MAX + cd.w;
        int keyv  = valid ? mykey : -1;                    // packed key+valid

        // A operand: per K-chunk c, this lane holds feats (4c+2*hi, 4c+2*hi+1)
        const float* fr = feat + (size_t)pc * C_IN + 2 * hi;
        v2f a0 = *(const v2f*)(fr + 0);
        v2f a1 = *(const v2f*)(fr + 4);
        v2f a2 = *(const v2f*)(fr + 8);
        v2f a3 = *(const v2f*)(fr + 12);

        v8f acc0 = {0.f, 0.f, 0.f, 0.f, 0.f, 0.f, 0.f, 0.f};  // out cols 0..15
        v8f acc1 = {0.f, 0.f, 0.f, 0.f, 0.f, 0.f, 0.f, 0.f};  // out cols 16..31

        unsigned rem = __builtin_amdgcn_ballot_w32(valid != 0) & 0xFFFFu;
        while (rem) {                                   // distinct kernel ids
            int lead = __builtin_ctz(rem);
            int kk = __builtin_amdgcn_readlane(myk, lead);
            bool match = (myk == kk) && (valid != 0);
            float msk = match ? 1.0f : 0.0f;            // row-mask A (no EXEC div.)
            const float* bk = sWT + kk * 512;
#pragma unroll
            for (int c = 0; c < 4; ++c) {
                int i0 = 4 * c + 2 * hi;
                v2f ac = (c == 0) ? a0 : (c == 1) ? a1 : (c == 2) ? a2 : a3;
                v2f av; av.x = ac.x * msk; av.y = ac.y * msk;
                v2f b0 = *(const v2f*)(bk + n16 * 16 + i0);          // o = n16
                v2f b1 = *(const v2f*)(bk + (n16 + 16) * 16 + i0);   // o = n16+16
                acc0 = __builtin_amdgcn_wmma_f32_16x16x4_f32(
                    false, av, false, b0, (short)0, acc0, false, false);
                acc1 = __builtin_amdgcn_wmma_f32_16x16x4_f32(
                    false, av, false, b1, (short)0, acc1, false, false);
            }
            unsigned mm = __builtin_amdgcn_ballot_w32(match);
            rem &= ~mm;                                 // leader always clears
        }

        if (hi == 0 && valid) flags[mykey] = 1;         // occupancy (benign race)

        // Scatter D into dense grid: VGPR g holds row (g + 8*hi), column n16.
#pragma unroll
        for (int g = 0; g < 8; ++g) {
            int keyr = __shfl(keyv, g + 8 * hi, 32);    // ds_bpermute
            if (keyr >= 0) {
                atomicAdd(acc_grid + (size_t)keyr * C_OUT + n16,      acc0[g]);
                atomicAdd(acc_grid + (size_t)keyr * C_OUT + 16 + n16, acc1[g]);
            }
        }
    }
}

// ---------------------------------------------------------------------------
// Kernel 2: per-block (256 keys) occupancy counts via wave32 ballot/popc.
// ---------------------------------------------------------------------------
__global__ __launch_bounds__(256) void scan_blocks(
    const int* __restrict__ flags, int* __restrict__ bsums)
{
    __shared__ int ws[8];
    int i = blockIdx.x * 256 + threadIdx.x;
    int f = (flags[i] != 0);
    unsigned mask = __builtin_amdgcn_ballot_w32(f != 0);
    if ((threadIdx.x & 31) == 0) ws[threadIdx.x >> 5] = __popc(mask);
    __syncthreads();
    if (threadIdx.x == 0) {
        int s = 0;
#pragma unroll
        for (int w = 0; w < 8; ++w) s += ws[w];
        bsums[blockIdx.x] = s;
    }
}

// ---------------------------------------------------------------------------
// Kernel 3: exclusive scan of the 1024 block sums (one block, Hillis-Steele).
// ---------------------------------------------------------------------------
__global__ __launch_bounds__(1024) void scan_top(int* __restrict__ bsums)
{
    __shared__ int sc[1024];
    int t = threadIdx.x;
    int v = bsums[t];
    sc[t] = v;
    __syncthreads();
    for (int off = 1; off < 1024; off <<= 1) {
        int add = (t >= off) ? sc[t - off] : 0;
        __syncthreads();
        sc[t] += add;
        __syncthreads();
    }
    bsums[t] = sc[t] - v;   // exclusive
}

// ---------------------------------------------------------------------------
// Kernel 4: compact occupied keys in ascending order (== jnp.unique order):
// rank = block base + wave base + intra-wave ballot prefix; copy 32 channels
// and decode coords (x, 0, y, batch).
// ---------------------------------------------------------------------------
__global__ __launch_bounds__(256) void compact_kernel(
    const int* __restrict__ flags, const int* __restrict__ bsums,
    const float* __restrict__ acc_grid,
    float* __restrict__ outf, int* __restrict__ outc, int n)
{
    __shared__ int ws[8];
    int i = blockIdx.x * 256 + threadIdx.x;
    int f = (flags[i] != 0);
    unsigned mask = __builtin_amdgcn_ballot_w32(f != 0);
    int lane = threadIdx.x & 31;
    int warp = threadIdx.x >> 5;
    if (lane == 0) ws[warp] = __popc(mask);
    __syncthreads();
    int wb = 0;
    for (int w = 0; w < warp; ++w) wb += ws[w];
    int prefix = __popc(mask & ((1u << lane) - 1u));
    int rank = bsums[blockIdx.x] + wb + prefix;
    if (f && rank < n) {
        outc[(size_t)rank * 4 + 0] = i >> 12;          // x
        outc[(size_t)rank * 4 + 1] = 0;                // z flattened
        outc[(size_t)rank * 4 + 2] = (i >> 6) & 63;    // y
        outc[(size_t)rank * 4 + 3] = i & 63;           // batch
        const v4f* src = (const v4f*)(acc_grid + (size_t)i * C_OUT);
        v4f* dst = (v4f*)(outf + (size_t)rank * C_OUT);
#pragma unroll
        for (int j = 0; j < 8; ++j) dst[j] = src[j];
    }
}

// ---------------------------------------------------------------------------
extern "C" void kernel_launch(void* const* d_in, const int* in_sizes, int n_in,
                              void* d_out, int out_size, void* d_ws, size_t ws_size,
                              hipStream_t stream)
{
    const float* feat    = (const float*)d_in[0];
    const int*   coords  = (const int*)d_in[1];
    const float* W       = (const float*)d_in[2];
    const int*   strideP = (const int*)d_in[3];
    const int n = in_sizes[0] / C_IN;

    float* outf = (float*)d_out;
    int*   outc = (int*)(outf + (size_t)n * C_OUT);

    float* acc_grid = (float*)d_ws;                                   // 32 MB
    int*   flags = (int*)((char*)d_ws + (size_t)KEYS * C_OUT * sizeof(float));
    int*   bsums = (int*)((char*)flags + (size_t)KEYS * sizeof(int)); // 4 KB

    zero_init_kernel<<<4096, 256, 0, stream>>>(acc_grid, flags, outf, outc,
                                               (long long)n);
    bev_gemv_scatter<<<1024, 256, 0, stream>>>(feat, coords, W, strideP,
                                               acc_grid, flags, n);
    scan_blocks<<<KEYS / 256, 256, 0, stream>>>(flags, bsums);
    scan_top<<<1, 1024, 0, stream>>>(bsums);
    compact_kernel<<<KEYS / 256, 256, 0, stream>>>(flags, bsums, acc_grid,
                                                   outf, outc, n);
}